// SpikesEncoder_18528488915114
// MI455X (gfx1250) — compile-verified
//
#include <hip/hip_runtime.h>

// LIF spiking network forward on MI455X (gfx1250, wave32).
//
// Phase 1: i_in = x @ W   (M=256, N=2048, K=2048, fp32)  -> V_WMMA_F32_16X16X4_F32
// Phase 2: 100-step LIF recurrence per element, spikes streamed to [B,T,N].
//
// Roofline: output = 256*100*2048*4B = 209.7 MB of stores -> ~9 us at 23.3 TB/s.
// GEMM is 2.15 GFLOP (negligible); x (2MB) + W (16MB) live in the 192MB L2.
// => fuse everything into one store-bandwidth-bound kernel, keep fp32 precision.

typedef __attribute__((ext_vector_type(2))) float v2f;
typedef __attribute__((ext_vector_type(8))) float v8f;

#define LIF_ALPHA 0.90483741803595952f  // exp(-dt/tau_m) = exp(-0.1)
#define T_STEPS   100
#define N_BATCH   256
#define N_UNITS   2048
#define N_TILES_N (N_UNITS / 16)        // 128

__global__ __launch_bounds__(32)
void lif_wmma_fused_kernel(const float* __restrict__ x,
                           const float* __restrict__ w,
                           float* __restrict__ out)
{
    const int lane = threadIdx.x & 31;
    const int lo   = lane & 15;   // column / M-row selector within half-wave
    const int hi   = lane >> 4;   // which half of the wave

    // One wave per 16x16 tile of i_in.
    const int tile  = blockIdx.x;            // 0 .. 16*128-1
    const int tileM = tile >> 7;             // / N_TILES_N
    const int tileN = tile & (N_TILES_N - 1);
    const int tm0   = tileM * 16;
    const int tn0   = tileN * 16;

    // ---------------- Phase 1: fp32 WMMA GEMM, K loop in steps of 4 ----------
    // A fragment (16x4, x): lane holds A[lo][k + 2*hi + {0,1}]  (float2 load)
    // B fragment (4x16, W): lane holds B[k + 2*hi + {0,1}][tn0+lo]
    v8f acc = {0.f, 0.f, 0.f, 0.f, 0.f, 0.f, 0.f, 0.f};

    const float* aptr = x + (size_t)(tm0 + lo) * N_UNITS + 2 * hi;
    const float* bptr = w + (size_t)(2 * hi) * N_UNITS + tn0 + lo;

    #pragma unroll 4
    for (int k = 0; k < N_UNITS; k += 4) {
        v2f a = *(const v2f*)aptr;           // 8B aligned: 2*hi is even
        v2f b;
        b.x = bptr[0];
        b.y = bptr[N_UNITS];
        // D = A*B + C  (8 args: neg_a, A, neg_b, B, c_mod, C, reuse_a, reuse_b)
        acc = __builtin_amdgcn_wmma_f32_16x16x4_f32(
            /*neg_a=*/false, a, /*neg_b=*/false, b,
            /*c_mod=*/(short)0, acc, /*reuse_a=*/false, /*reuse_b=*/false);
        aptr += 4;
        bptr += 4 * N_UNITS;
    }

    // ---------------- Phase 2: LIF recurrence on the accumulator -------------
    // C/D layout: VGPR r, lane L -> element (M = r + 8*(L>>4), N = L&15).
    // Each lane owns 8 i_in values; run the 100-step loop in registers and
    // stream spikes to out[b][t][n].
    float V[8], Z[8];
    size_t base[8];
    #pragma unroll
    for (int r = 0; r < 8; ++r) {
        V[r] = 0.f;
        Z[r] = 0.f;
        const int b = tm0 + r + 8 * hi;                 // batch row
        base[r] = (size_t)b * T_STEPS * N_UNITS + tn0 + lo;
    }

    for (int t = 0; t < T_STEPS; ++t) {
        #pragma unroll
        for (int r = 0; r < 8; ++r) {
            const float i_in = acc[r];
            // V' = alpha*V*(1-Z) + i_in with Z in {0,1}:
            const float v = (Z[r] != 0.f) ? i_in : fmaf(LIF_ALPHA, V[r], i_in);
            const float z = (v >= 1.0f) ? 1.0f : 0.0f;
            V[r] = v;
            Z[r] = z;
            out[base[r]] = z;       // contiguous 16-float rows per half-wave
            base[r] += N_UNITS;     // advance one timestep
        }
    }
}

extern "C" void kernel_launch(void* const* d_in, const int* in_sizes, int n_in,
                              void* d_out, int out_size, void* d_ws, size_t ws_size,
                              hipStream_t stream) {
    const float* x = (const float*)d_in[0];               // [256, 2048]
    const float* w = (const float*)d_in[1];               // [2048, 2048]
    float* out = (float*)d_out;                           // [256, 100, 2048]
    (void)in_sizes; (void)n_in; (void)out_size; (void)d_ws; (void)ws_size;

    const int n_tiles = (N_BATCH / 16) * N_TILES_N;       // 16 * 128 = 2048 waves
    lif_wmma_fused_kernel<<<dim3(n_tiles), dim3(32), 0, stream>>>(x, w, out);
}